// DynamicConv2d_3728031613529
// MI455X (gfx1250) — compile-verified
//
#include <hip/hip_runtime.h>
#include <hip/hip_bf16.h>

typedef __attribute__((ext_vector_type(2))) float v2f;
typedef __attribute__((ext_vector_type(4))) float v4f;
typedef __attribute__((ext_vector_type(8))) float v8f;

#define Bsz 16
#define Cch 128
#define Hh 64
#define Ww 64
#define Och 128
#define Kk 4
#define Rr 32   // C/4

// ---- workspace layout (floats) ----
// [0)                      W_eff transposed: [b][tap(9)][o(128)][c(128)]  = 2,359,296 floats
// [WS_POOLED)              pooled: [b][c]                                 = 2,048 floats
// [WS_ATTN)                attn:   [b][k]                                 = 64 floats
#define WS_WEFF   0
#define WS_POOLED (16*9*128*128)
#define WS_ATTN   (WS_POOLED + Bsz*Cch)

// ---------------------------------------------------------------------------
// Kernel 1: global average pool.  One block per (b,c); 256 threads reduce 4096.
// ---------------------------------------------------------------------------
__global__ void k_pool(const float* __restrict__ x, float* __restrict__ pooled) {
    __shared__ float red[256];
    const int bc  = blockIdx.x;          // b*C + c
    const int tid = threadIdx.x;
    const float* p = x + (size_t)bc * (Hh * Ww);
    float s = 0.f;
    for (int i = tid; i < Hh * Ww; i += 256) s += p[i];
    red[tid] = s;
    __syncthreads();
    for (int off = 128; off > 0; off >>= 1) {
        if (tid < off) red[tid] += red[tid + off];
        __syncthreads();
    }
    if (tid == 0) pooled[bc] = red[0] * (1.0f / (Hh * Ww));
}

// ---------------------------------------------------------------------------
// Kernel 2: SE head. One wave32 block per batch: h = relu(pooled@w1^T + b1),
// attn = softmax(h@w2^T + b2).
// ---------------------------------------------------------------------------
__global__ void k_se_head(const float* __restrict__ pooled,
                          const float* __restrict__ w1, const float* __restrict__ b1,
                          const float* __restrict__ w2, const float* __restrict__ b2,
                          float* __restrict__ attn) {
    __shared__ float sh[Rr];
    __shared__ float sl[Kk];
    const int b = blockIdx.x;
    const int t = threadIdx.x;           // 0..31

    // hidden layer: r = t
    {
        const float* pv = pooled + b * Cch;
        const float* wr = w1 + t * Cch;
        float acc = b1[t];
        for (int c = 0; c < Cch; ++c) acc += pv[c] * wr[c];
        sh[t] = acc > 0.f ? acc : 0.f;
    }
    __syncthreads();
    if (t < Kk) {
        const float* wr = w2 + t * Rr;
        float acc = b2[t];
        for (int r = 0; r < Rr; ++r) acc += sh[r] * wr[r];
        sl[t] = acc;
    }
    __syncthreads();
    if (t < Kk) {
        float m = sl[0];
        for (int k = 1; k < Kk; ++k) m = fmaxf(m, sl[k]);
        float den = 0.f;
        for (int k = 0; k < Kk; ++k) den += __expf(sl[k] - m);
        attn[b * Kk + t] = __expf(sl[t] - m) / den;
    }
}

// ---------------------------------------------------------------------------
// Kernel 3: fold attention into the kernel bank.
//   Weff_t[b][tap][o][c] = sum_k attn[b][k] * kernels[k][o][c][tap]
// Tap-major layout makes the conv kernel's per-tap A-slice contiguous (64 KB).
// grid = (O*C/256, B); one thread per (o,c).
// ---------------------------------------------------------------------------
__global__ void k_fuse(const float* __restrict__ kernels,
                       const float* __restrict__ attn,
                       float* __restrict__ weff) {
    const int b   = blockIdx.y;
    const int idx = blockIdx.x * 256 + threadIdx.x;    // o*128 + c
    const int o   = idx >> 7;
    const int c   = idx & 127;

    float a0 = attn[b * Kk + 0], a1 = attn[b * Kk + 1];
    float a2 = attn[b * Kk + 2], a3 = attn[b * Kk + 3];

    const size_t kstride = (size_t)Och * Cch * 9;       // between k's
    const float* kp = kernels + ((size_t)o * Cch + c) * 9;

    float* out = weff + (size_t)b * 9 * Och * Cch;
#pragma unroll
    for (int tap = 0; tap < 9; ++tap) {
        float v = a0 * kp[tap]
                + a1 * kp[kstride + tap]
                + a2 * kp[2 * kstride + tap]
                + a3 * kp[3 * kstride + tap];
        out[((size_t)tap * Och + o) * Cch + c] = v;
    }
}

// ---------------------------------------------------------------------------
// Kernel 4: implicit-GEMM conv with per-sample fused weights, f32 WMMA.
//   M = O = 128 (8 waves x 16), N = 64 (one output row, 4 subtiles of 16),
//   K = C*9 stepped 4 at a time via v_wmma_f32_16x16x4_f32.
// grid = (H, B), 256 threads (8 wave32), dynamic LDS = 100 KB.
// LDS: A[128][132] (stride 132 == 4 mod 64 -> conflict-free column reads),
//      X[128][68]  (haloed input row, 66 valid + pad).
// ---------------------------------------------------------------------------
#define A_STRIDE 132
#define X_STRIDE 68
#define A_FLOATS (Cch * A_STRIDE)

__global__ __launch_bounds__(256)
void k_conv(const float* __restrict__ x,
            const float* __restrict__ weff,
            float* __restrict__ out) {
    extern __shared__ float smem[];
    float* As = smem;                 // 128 x 132
    float* Xs = smem + A_FLOATS;      // 128 x 68 (66 valid)

    const int h    = blockIdx.x;
    const int b    = blockIdx.y;
    const int tid  = threadIdx.x;
    const int wv   = tid >> 5;        // 0..7  -> M-tile (output channels wv*16..)
    const int lane = tid & 31;
    const int row  = lane & 15;       // M (or N) index within tile
    const int hi   = lane >> 4;       // upper half-wave -> K+2 / M+8

    const float* wbase = weff + (size_t)b * 9 * Och * Cch;

    v8f acc[4] = {};                  // four 16x16 N-subtiles

    for (int kh = 0; kh < 3; ++kh) {
        const int  hin = h + kh - 1;
        const bool hok = (hin >= 0) && (hin < Hh);
        __syncthreads();              // previous kh finished reading Xs
        // stage haloed input row: Xs[c][j], j=0..65 maps to win=j-1
        {
            const float* xrow = x + (((size_t)(b * Cch) * Hh) + (hok ? hin : 0)) * Ww;
            for (int i = tid; i < Cch * 66; i += 256) {
                const int c   = i / 66;
                const int j   = i - c * 66;
                const int win = j - 1;
                float v = 0.f;
                if (hok && win >= 0 && win < Ww)
                    v = xrow[(size_t)c * (Hh * Ww) + win];
                Xs[c * X_STRIDE + j] = v;
            }
        }
        for (int kw = 0; kw < 3; ++kw) {
            const int tap = kh * 3 + kw;
            __syncthreads();          // previous tap finished reading As; Xs ready
            // stage W_eff tap slice [o][c] (contiguous 64 KB) via float4
            {
                const v4f* src = (const v4f*)(wbase + (size_t)tap * Och * Cch);
                for (int i = tid; i < (Och * Cch) / 4; i += 256) {
                    v4f v = src[i];
                    const int o = (i * 4) >> 7;
                    const int c = (i * 4) & 127;
                    *(v4f*)&As[o * A_STRIDE + c] = v;
                }
                // hint the next tap's slice toward L2 (global_prefetch_b8)
                if (tap < 8 && tid == 0)
                    __builtin_prefetch(wbase + (size_t)(tap + 1) * Och * Cch, 0, 1);
            }
            __syncthreads();          // As + Xs visible

            const int m0 = wv * 16;
#pragma unroll 4
            for (int cc = 0; cc < Cch; cc += 4) {
                // A fragment (16x4): lane&15 = M, upper half-wave holds K+2
                v2f a;
                const float* ap = &As[(m0 + row) * A_STRIDE + cc + 2 * hi];
                a.x = ap[0];
                a.y = ap[1];
#pragma unroll
                for (int ns = 0; ns < 4; ++ns) {
                    // B fragment (4x16): lane&15 = N, upper half-wave holds K+2
                    v2f bb;
                    const float* bp = &Xs[(cc + 2 * hi) * X_STRIDE + ns * 16 + row + kw];
                    bb.x = bp[0];
                    bb.y = bp[X_STRIDE];
                    acc[ns] = __builtin_amdgcn_wmma_f32_16x16x4_f32(
                        false, a, false, bb, (short)0, acc[ns], false, false);
                }
            }
        }
    }

    // D layout: VGPR r -> M = r + 8*hi, N = lane&15
    float* obase = out + (((size_t)b * Och) * Hh + h) * Ww;
#pragma unroll
    for (int ns = 0; ns < 4; ++ns) {
#pragma unroll
        for (int r = 0; r < 8; ++r) {
            const int o = wv * 16 + r + 8 * hi;
            obase[(size_t)o * (Hh * Ww) + ns * 16 + row] = acc[ns][r];
        }
    }
}

// ---------------------------------------------------------------------------
extern "C" void kernel_launch(void* const* d_in, const int* in_sizes, int n_in,
                              void* d_out, int out_size, void* d_ws, size_t ws_size,
                              hipStream_t stream) {
    const float* x       = (const float*)d_in[0];
    const float* w1      = (const float*)d_in[1];
    const float* b1      = (const float*)d_in[2];
    const float* w2      = (const float*)d_in[3];
    const float* b2      = (const float*)d_in[4];
    const float* kernels = (const float*)d_in[5];
    float*       out     = (float*)d_out;
    float*       ws      = (float*)d_ws;

    float* weff   = ws + WS_WEFF;
    float* pooled = ws + WS_POOLED;
    float* attn   = ws + WS_ATTN;

    // 1) global average pool
    k_pool<<<dim3(Bsz * Cch), dim3(256), 0, stream>>>(x, pooled);
    // 2) SE head -> attention weights
    k_se_head<<<dim3(Bsz), dim3(32), 0, stream>>>(pooled, w1, b1, w2, b2, attn);
    // 3) fold attention into kernel bank (tap-major layout)
    k_fuse<<<dim3((Och * Cch) / 256, Bsz), dim3(256), 0, stream>>>(kernels, attn, weff);
    // 4) WMMA implicit-GEMM conv
    const size_t lds = (size_t)(A_FLOATS + Cch * X_STRIDE) * sizeof(float); // 100 KB
    k_conv<<<dim3(Hh, Bsz), dim3(256), lds, stream>>>(x, weff, out);
}